// SparsePolicy_61512521613576
// MI455X (gfx1250) — compile-verified
//
#include <hip/hip_runtime.h>
#include <hip/hip_bf16.h>

typedef __attribute__((ext_vector_type(2))) float v2f;
typedef __attribute__((ext_vector_type(8))) float v8f;

#define NODE_DIM 64
#define EDGE_DIM 32
#define HID_EDGE 4
#define KTOT     160      // 64 (xr) + 64 (xc) + 32 (edge_x)
#define NCHUNK   40       // KTOT / 4
#define LSEG     128      // LDS segment accumulator capacity (num_graphs = 100)
#define GSEG     1024     // global batch-sum capacity

// ---------------------------------------------------------------------------
// Precompute combined B matrix [KTOT x 4] (row-major, 4 cols) and fused bias.
//   cols 0,1 : attention weights; rows 128..159 are W_edge @ W_att[128:132,:]
//   cols 2,3 : W_thr on the xr rows (0..63), zero elsewhere
// ---------------------------------------------------------------------------
__global__ void prep_kernel(const float* __restrict__ W_edge,
                            const float* __restrict__ b_edge,
                            const float* __restrict__ W_att,
                            const float* __restrict__ b_att,
                            const float* __restrict__ W_thr,
                            const float* __restrict__ b_thr,
                            float* __restrict__ Bc, float* __restrict__ bias) {
    int k = blockIdx.x * blockDim.x + threadIdx.x;
    if (k < KTOT) {
        float c0, c1, c2, c3;
        if (k < 2 * NODE_DIM) {
            c0 = W_att[k * 2 + 0];
            c1 = W_att[k * 2 + 1];
        } else {
            int r = k - 2 * NODE_DIM;  // edge_x feature row
            c0 = 0.f; c1 = 0.f;
            for (int h = 0; h < HID_EDGE; ++h) {
                float we = W_edge[r * HID_EDGE + h];
                c0 += we * W_att[(2 * NODE_DIM + h) * 2 + 0];
                c1 += we * W_att[(2 * NODE_DIM + h) * 2 + 1];
            }
        }
        if (k < NODE_DIM) { c2 = W_thr[k * 2 + 0]; c3 = W_thr[k * 2 + 1]; }
        else              { c2 = 0.f; c3 = 0.f; }
        Bc[k * 4 + 0] = c0; Bc[k * 4 + 1] = c1;
        Bc[k * 4 + 2] = c2; Bc[k * 4 + 3] = c3;
    }
    if (k == 0) {
        float bb0 = b_att[0], bb1 = b_att[1];
        for (int h = 0; h < HID_EDGE; ++h) {
            bb0 += b_edge[h] * W_att[(2 * NODE_DIM + h) * 2 + 0];
            bb1 += b_edge[h] * W_att[(2 * NODE_DIM + h) * 2 + 1];
        }
        bias[0] = bb0; bias[1] = bb1;
        bias[2] = b_thr[0]; bias[3] = b_thr[1];
    }
}

// ---------------------------------------------------------------------------
// Main kernel: per 16-edge tile, 40-chunk V_WMMA_F32_16X16X4_F32 chain
// producing 4 logits per edge; epilogue computes att = relu(sig - sig),
// stores att[E,2] and accumulates per-graph sums.
// ---------------------------------------------------------------------------
__global__ __launch_bounds__(256) void edge_att_kernel(
    const float* __restrict__ x, const float* __restrict__ edge_x,
    const int* __restrict__ ei, const float* __restrict__ Bc,
    const float* __restrict__ bias, const int* __restrict__ pgp,
    float* __restrict__ att_out, float* __restrict__ bsum,
    int E, int nTiles, int nIter)
{
    __shared__ float segAcc[LSEG * 2];
    __shared__ float logits[8][16][4];   // 8 waves per block

    for (int i = threadIdx.x; i < LSEG * 2; i += blockDim.x) segAcc[i] = 0.f;
    __syncthreads();

    const int pg   = pgp[0];
    const int lane = threadIdx.x & 31;
    const int wave = threadIdx.x >> 5;
    const int L    = lane & 15;
    const int half = lane >> 4;          // 0: K+{0,1}, 1: K+{2,3}
    const int gwave      = blockIdx.x * 8 + wave;
    const int waveStride = gridDim.x * 8;

    // Register-resident B fragments (tile-invariant): 40 x v2f = 80 VGPRs.
    v2f Bv[NCHUNK];
#pragma unroll
    for (int kc = 0; kc < NCHUNK; ++kc) {
        int k0 = 4 * kc + 2 * half;
        if (L < 4) {
            Bv[kc][0] = Bc[k0 * 4 + L];
            Bv[kc][1] = Bc[(k0 + 1) * 4 + L];
        } else {
            Bv[kc][0] = 0.f; Bv[kc][1] = 0.f;
        }
    }
    const float b0 = bias[0], b1 = bias[1], b2 = bias[2], b3 = bias[3];

    for (int it = 0; it < nIter; ++it) {
        int  tile  = gwave + it * waveStride;
        int  e     = tile * 16 + L;
        bool valid = (tile < nTiles) && (e < E);
        int  ec    = valid ? e : (E - 1);          // clamp: EXEC stays all-1s

        int row = ei[ec];
        int col = ei[(long long)E + ec];
        const float* pr = x + (long long)row * NODE_DIM + 2 * half;
        const float* pc = x + (long long)col * NODE_DIM + 2 * half;
        const float* pe = edge_x + (long long)ec * EDGE_DIM + 2 * half;

        v8f acc = {};
#pragma unroll
        for (int kc = 0; kc < 16; ++kc) {          // xr rows, K = 0..63
            v2f a = *(const v2f*)(pr + 4 * kc);
            acc = __builtin_amdgcn_wmma_f32_16x16x4_f32(
                false, a, false, Bv[kc], (short)0, acc, false, false);
        }
#pragma unroll
        for (int kc = 0; kc < 16; ++kc) {          // xc rows, K = 64..127
            v2f a = *(const v2f*)(pc + 4 * kc);
            acc = __builtin_amdgcn_wmma_f32_16x16x4_f32(
                false, a, false, Bv[16 + kc], (short)0, acc, false, false);
        }
#pragma unroll
        for (int kc = 0; kc < 8; ++kc) {           // edge_x rows, K = 128..159
            v2f a = *(const v2f*)(pe + 4 * kc);
            acc = __builtin_amdgcn_wmma_f32_16x16x4_f32(
                false, a, false, Bv[32 + kc], (short)0, acc, false, false);
        }

        // D layout: lanes 0-15 hold M=v (vgpr v), lanes 16-31 hold M=v+8; N=lane&15.
        if (L < 4) {
#pragma unroll
            for (int v = 0; v < 8; ++v)
                logits[wave][half * 8 + v][L] = acc[v];
        }
        __syncthreads();

        if (lane < 16 && valid) {
            float l0 = logits[wave][L][0] + b0;   // att logit col 0
            float l1 = logits[wave][L][1] + b1;   // att logit col 1
            float l2 = logits[wave][L][2] + b2;   // thr logit col 0
            float l3 = logits[wave][L][3] + b3;   // thr logit col 1
            float s0 = 1.f / (1.f + __expf(-l0));
            float s1 = 1.f / (1.f + __expf(-l1));
            float t0 = 1.f / (1.f + __expf(-l2));
            float t1 = 1.f / (1.f + __expf(-l3));
            float a0 = fmaxf(s0 - t0, 0.f);
            float a1 = fmaxf(s1 - t1, 0.f);
            v2f av; av[0] = a0; av[1] = a1;
            *(v2f*)(att_out + (long long)e * 2) = av;

            int b = (pg > 0) ? (row / pg) : 0;
            if (b < LSEG) {
                atomicAdd(&segAcc[b * 2 + 0], a0);
                atomicAdd(&segAcc[b * 2 + 1], a1);
            } else {
                b = b < GSEG ? b : (GSEG - 1);
                atomicAdd(&bsum[b * 2 + 0], a0);
                atomicAdd(&bsum[b * 2 + 1], a1);
            }
        }
        __syncthreads();
    }

    for (int i = threadIdx.x; i < LSEG * 2; i += blockDim.x) {
        float v = segAcc[i];
        if (v != 0.f) atomicAdd(&bsum[i], v);
    }
}

// ---------------------------------------------------------------------------
// Finalize: out[e] = sum_n att[e,n] / (bsum[batch,n] + 1e-5) * aa[e,n]
// ---------------------------------------------------------------------------
__global__ __launch_bounds__(256) void finalize_kernel(
    const float* __restrict__ att, const float* __restrict__ aa,
    const int* __restrict__ ei, const int* __restrict__ pgp,
    const float* __restrict__ bsum, float* __restrict__ out, int E)
{
    int e = blockIdx.x * blockDim.x + threadIdx.x;
    if (e >= E) return;
    int pg  = pgp[0];
    int row = ei[e];
    int b   = (pg > 0) ? (row / pg) : 0;
    b = b < GSEG ? b : (GSEG - 1);
    float s0 = bsum[b * 2 + 0] + 1e-5f;
    float s1 = bsum[b * 2 + 1] + 1e-5f;
    v2f a = *(const v2f*)(att + (long long)e * 2);
    v2f m = *(const v2f*)(aa + (long long)e * 2);
    out[e] = a[0] / s0 * m[0] + a[1] / s1 * m[1];
}

extern "C" void kernel_launch(void* const* d_in, const int* in_sizes, int n_in,
                              void* d_out, int out_size, void* d_ws, size_t ws_size,
                              hipStream_t stream) {
    (void)n_in; (void)out_size; (void)ws_size;
    const float* x       = (const float*)d_in[0];
    const float* edge_x  = (const float*)d_in[1];
    const int*   ei      = (const int*)d_in[2];
    const float* aa      = (const float*)d_in[3];
    const int*   pgp     = (const int*)d_in[4];
    const float* W_edge  = (const float*)d_in[5];
    const float* b_edge  = (const float*)d_in[6];
    const float* W_att   = (const float*)d_in[7];
    const float* b_att   = (const float*)d_in[8];
    const float* W_thr   = (const float*)d_in[9];
    const float* b_thr   = (const float*)d_in[10];

    int E = in_sizes[3] / 2;   // actual_amount is [E,2]

    char*  ws     = (char*)d_ws;
    float* bsum   = (float*)ws;                              // 8192 B
    float* Bc     = (float*)(ws + 8192);                     // 2560 B
    float* bias   = (float*)(ws + 8192 + 2560);              // 16 B
    float* attbuf = (float*)(ws + 8192 + 2560 + 16);         // E*8 B

    hipMemsetAsync(bsum, 0, GSEG * 2 * sizeof(float), stream);
    prep_kernel<<<1, 256, 0, stream>>>(W_edge, b_edge, W_att, b_att,
                                       W_thr, b_thr, Bc, bias);

    int nTiles = (E + 15) / 16;
    int blocks = 2048;
    long long totalWaves = (long long)blocks * 8;
    if (totalWaves > nTiles) {
        blocks = (nTiles + 7) / 8;
        if (blocks < 1) blocks = 1;
        totalWaves = (long long)blocks * 8;
    }
    int nIter = (int)((nTiles + totalWaves - 1) / totalWaves);

    edge_att_kernel<<<blocks, 256, 0, stream>>>(
        x, edge_x, ei, Bc, bias, pgp, attbuf, bsum, E, nTiles, nIter);

    finalize_kernel<<<(E + 255) / 256, 256, 0, stream>>>(
        attbuf, aa, ei, pgp, bsum, (float*)d_out, E);
}